// BalanceLoss_17987323036123
// MI455X (gfx1250) — compile-verified
//
#include <hip/hip_runtime.h>
#include <cmath>

// Problem constants (match setup_inputs)
#define B_ROWS 131072
#define C_CH   128
#define N_ELEM (B_ROWS * C_CH)   // 16777216
#define N_F4   (N_ELEM / 4)      // 4194304 float4 elements
#define GRID   512
#define BLOCK  256               // 8 waves (wave32)

typedef __attribute__((ext_vector_type(2))) float v2f;
typedef __attribute__((ext_vector_type(8))) float v8f;

// Workspace layout (floats):
//  [0..127]    pos_sum per channel (atomic, exact integer-valued f32 -> deterministic)
//  [128..255]  maj_scale per channel
//  [256..383]  min_scale per channel
//  [384..511]  pos_gt    per channel
//  [512..1023] per-block partial loss sums (non-atomic -> deterministic)

__global__ void bl_init(float* __restrict__ ws) {
    int t = threadIdx.x;
    if (t < C_CH) ws[t] = 0.0f;
}

// Phase 1: pos_sum[c] = sum_b target[b,c].  target is {0,1}; all partial sums
// are integers < 2^24, so f32 atomic adds are exact and order-independent.
// float4 loads: stride (GRID*BLOCK f4) is a multiple of 32 f4 = 128 floats,
// so each thread's channel group 4*(t&31) is loop-invariant.
__global__ void bl_colsum(const float4* __restrict__ target4, float* __restrict__ ws) {
    int t = threadIdx.x;
    int i = blockIdx.x * BLOCK + t;
    const int stride = GRID * BLOCK;          // N_F4 / stride = 32 exact iterations
    float4 acc = make_float4(0.f, 0.f, 0.f, 0.f);
    #pragma unroll 4
    for (; i < N_F4; i += stride) {
        float4 v = target4[i];
        acc.x += v.x; acc.y += v.y; acc.z += v.z; acc.w += v.w;
    }
    __shared__ float4 s[BLOCK];
    s[t] = acc;
    __syncthreads();
    if (t < 32) {
        float4 v = s[t];
        #pragma unroll
        for (int w = 1; w < 8; ++w) {
            float4 u = s[t + 32 * w];
            v.x += u.x; v.y += u.y; v.z += u.z; v.w += u.w;
        }
        int c = t * 4;                         // channel group of lane t
        atomicAdd(&ws[c + 0], v.x);
        atomicAdd(&ws[c + 1], v.y);
        atomicAdd(&ws[c + 2], v.z);
        atomicAdd(&ws[c + 3], v.w);
    }
}

// Tiny per-channel statistics.  maj_cnt/min_cnt are derivable from pos_sum
// because target, pos_gt, neg_gt are all {0,1}.
__global__ void bl_scales(float* __restrict__ ws) {
    int c = threadIdx.x;                       // 128 threads
    const float Bf  = (float)B_ROWS;
    const float bal = 0.5f * Bf;               // POS_PROP * B
    float ps = ws[c];
    float ns = Bf - ps;
    float pgt = (ps >= bal) ? 1.0f : 0.0f;
    float ngt = (ns >  bal) ? 1.0f : 0.0f;
    float maj_cnt = (pgt == 1.0f) ? ps : ns;   // #(target == pos_gt)
    float min_cnt = (ngt == 1.0f) ? ps : ns;   // #(target == neg_gt)
    float maj_scale = (maj_cnt > 0.0f) ? (bal / fmaxf(maj_cnt, 1.0f)) : 1.0f;
    float min_scale = (min_cnt > 0.0f) ? ((Bf - bal) / fmaxf(min_cnt, 1.0f)) : 1.0f;
    ws[128 + c] = maj_scale;
    ws[256 + c] = min_scale;
    ws[384 + c] = pgt;
}

// Per-element weighted BCE (stable): softplus(-x) + (1-t)*x, weight per the
// balance rule: maj ? (easy ? 0 : maj_scale) : min_scale  (min_mask == !maj).
__device__ __forceinline__ float bl_elem(float x, float tg, float wmaj, float wmin, float pgt) {
    const float EASY_HI = (float)(10.0 / 30.0);     // EDGES[10]; g >= EDGES[0]=0 always
    float ax  = fabsf(x);
    float em  = __expf(-ax);                        // exp(-|x|) in (0,1]
    float inv = __builtin_amdgcn_rcpf(1.0f + em);
    float sig = (x >= 0.0f) ? inv : em * inv;       // sigmoid(x), stable
    float g   = fabsf(sig - tg);
    bool  maj = (tg == pgt);
    float w   = maj ? ((g < EASY_HI) ? 0.0f : wmaj) : wmin;
    float sp  = fmaxf(-x, 0.0f) + __logf(1.0f + em);  // softplus(-x), stable
    float bce = sp + (1.0f - tg) * x;
    return bce * w;
}

// Phase 2: weighted BCE sum.  Memory-bound: streams pred from HBM, target from
// L2 (134 MB total < 192 MB L2), b128 loads.  Wave reduction via
// V_WMMA_F32_16X16X4_F32 with an all-ones A matrix: the B operand {acc, 0} is a
// bijective fill of the 4x16 B matrix, so D row 0 (lanes 0..15 of D VGPR0) sums
// to the 32-lane total regardless of the exact B element mapping.
__global__ void bl_loss(const float4* __restrict__ pred4,
                        const float4* __restrict__ targ4,
                        const float* __restrict__ ws,
                        float* __restrict__ wsout) {
    int t  = threadIdx.x;
    int cg = (t & 31) * 4;                     // loop-invariant channel group
    const float4 wmaj = *(const float4*)&ws[128 + cg];
    const float4 wmin = *(const float4*)&ws[256 + cg];
    const float4 pgt  = *(const float4*)&ws[384 + cg];
    int i = blockIdx.x * BLOCK + t;
    const int stride = GRID * BLOCK;           // 32 exact iterations
    float4 acc = make_float4(0.f, 0.f, 0.f, 0.f);
    #pragma unroll 2
    for (; i < N_F4; i += stride) {
        float4 x  = pred4[i];
        float4 tg = targ4[i];
        acc.x += bl_elem(x.x, tg.x, wmaj.x, wmin.x, pgt.x);
        acc.y += bl_elem(x.y, tg.y, wmaj.y, wmin.y, pgt.y);
        acc.z += bl_elem(x.z, tg.z, wmaj.z, wmin.z, pgt.z);
        acc.w += bl_elem(x.w, tg.w, wmaj.w, wmin.w, pgt.w);
    }
    float lacc = (acc.x + acc.y) + (acc.z + acc.w);

    // Wave32 reduction: ones-A WMMA collapses 32 lane partials.
    v2f a; a.x = 1.0f; a.y = 1.0f;   // A = all ones (layout irrelevant)
    v2f b; b.x = lacc; b.y = 0.0f;   // B = lane partials + zeros
    v8f c = {};
    v8f d = __builtin_amdgcn_wmma_f32_16x16x4_f32(
        /*neg_a=*/false, a, /*neg_b=*/false, b,
        /*c_mod=*/(short)0, c, /*reuse_a=*/false, /*reuse_b=*/false);
    float r = d[0];                  // lane n (0..15): column-sum n of B
    r += __shfl_xor(r, 8, 32);
    r += __shfl_xor(r, 4, 32);
    r += __shfl_xor(r, 2, 32);
    r += __shfl_xor(r, 1, 32);       // lane 0: sum of lanes 0..15 = wave total
    __shared__ float sw[BLOCK / 32];
    int lane = t & 31, wid = t >> 5;
    if (lane == 0) sw[wid] = r;
    __syncthreads();
    if (t == 0) {
        float tot = 0.0f;
        #pragma unroll
        for (int i2 = 0; i2 < BLOCK / 32; ++i2) tot += sw[i2];
        wsout[512 + blockIdx.x] = tot;  // non-atomic -> deterministic
    }
}

__global__ void bl_final(const float* __restrict__ ws, float* __restrict__ out) {
    __shared__ float s[GRID];
    int t = threadIdx.x;             // 512 threads
    s[t] = ws[512 + t];
    __syncthreads();
    for (int off = GRID / 2; off > 0; off >>= 1) {
        if (t < off) s[t] += s[t + off];
        __syncthreads();
    }
    if (t == 0) out[0] = s[0] * (1.0f / (float)N_ELEM);  // 2^-24, exact scale
}

extern "C" void kernel_launch(void* const* d_in, const int* in_sizes, int n_in,
                              void* d_out, int out_size, void* d_ws, size_t ws_size,
                              hipStream_t stream) {
    const float4* pred4 = (const float4*)d_in[0];
    const float4* targ4 = (const float4*)d_in[1];
    float* ws  = (float*)d_ws;
    float* out = (float*)d_out;
    bl_init  <<<1,    256,   0, stream>>>(ws);
    bl_colsum<<<GRID, BLOCK, 0, stream>>>(targ4, ws);
    bl_scales<<<1,    128,   0, stream>>>(ws);
    bl_loss  <<<GRID, BLOCK, 0, stream>>>(pred4, targ4, ws, ws);
    bl_final <<<1,    GRID,  0, stream>>>(ws, out);
}